// Dyan_22728966931066
// MI455X (gfx1250) — compile-verified
//
#include <hip/hip_runtime.h>
#include <cmath>

// ---------------------------------------------------------------------------
// DYAN sparse coding on MI455X (gfx1250).
// Dominant cost: 30 FISTA iterations of DtD(513x513) @ C(513x512) x 32 batches
// => ~274 GFLOP (528-padded) f32 GEMM. Entire working set (~140 MB) is L2-resident
// (192 MB L2), so this is matrix-pipe bound: use V_WMMA_F32_16X16X4_F32
// (full fp32 accuracy, required for 30 chained FISTA iterations).
// Wave tile 16x64 (4 accumulators) + transposed B LDS layout so each WMMA
// operand K-pair is one contiguous 8B LDS read (no v_mov repacking).
// ---------------------------------------------------------------------------

typedef __attribute__((ext_vector_type(2))) float v2f;
typedef __attribute__((ext_vector_type(8))) float v8f;

#define T_FRAMES 64
#define NP_TRUE  513          // 2N+1
#define NP       528          // padded to multiple of 48 (=16*33, 528/48=11)
#define NJ       512
#define NBATCH   32
#define NITER    30
#define LAMBDA_C 0.1f
#define TOL2     1e-10f       // tol^2, tol = 1e-5

#define BM 48                 // block tile M (3 wave-rows of 16)
#define BN 128                // block tile N (2 wave-cols of 64)
#define KT 16                 // K tile staged through LDS
#define KP (KT + 2)           // LDS row stride: even => 8B-aligned K-pairs
#define GEMM_THREADS 192      // 6 waves (wave32)

// ------------------------- dictionary construction -------------------------
// One thread per padded column p of D (p in [0,528)); threads 528..543 idle.
// Column 0: ones; 1..256: r^i cos(i*theta); 257..512: r^i sin(i*theta);
// 513..527: zeros (padding). Also zeroes the per-launch scalar state.
__global__ void k_build_dict(const float* __restrict__ r,
                             const float* __restrict__ th,
                             float* __restrict__ Draw,
                             float* __restrict__ Dn,
                             float* __restrict__ scal) {
  int p = threadIdx.x;
  if (p == 0) {
    scal[0] = 0.f;                 // ||DtD||_F^2 accumulator
    *(int*)(scal + 3) = 0;         // sticky "frozen" flag
    for (int i = 0; i < NITER; ++i) scal[4 + i] = 0.f;  // per-iter norm^2
  }
  if (p >= NP) return;

  int mode;                        // 0=ones 1=cos 2=sin 3=zero-pad
  float rv = 0.f, tv = 0.f;
  if (p == 0)        { mode = 0; }
  else if (p <= 256) { mode = 1; rv = r[p - 1];   tv = th[p - 1]; }
  else if (p <= 512) { mode = 2; rv = r[p - 257]; tv = th[p - 257]; }
  else               { mode = 3; }

  float n2 = 0.f;
  for (int i = 0; i < T_FRAMES; ++i) {
    float v;
    if (mode == 0)      v = 1.f;
    else if (mode == 3) v = 0.f;
    else {
      float rp = powf(rv, (float)i);
      float a  = (float)i * tv;
      v = rp * ((mode == 1) ? cosf(a) : sinf(a));
    }
    Draw[i * NP + p] = v;
    n2 += v * v;
  }
  float inv = (p < NP_TRUE) ? (1.f / sqrtf(n2)) : 0.f;
  for (int i = 0; i < T_FRAMES; ++i)
    Dn[i * NP + p] = Draw[i * NP + p] * inv;
}

// ------------------------------- zero fill ---------------------------------
__global__ __launch_bounds__(256) void k_zero4(float4* __restrict__ p, long n4) {
  long i = (long)blockIdx.x * 256 + threadIdx.x;
  if (i < n4) p[i] = float4{0.f, 0.f, 0.f, 0.f};
}

// ------------------------ DtD + Frobenius norm^2 ---------------------------
__global__ __launch_bounds__(256)
void k_dtd(const float* __restrict__ Dn, float* __restrict__ DtD,
           float* __restrict__ scal) {
  int tid = blockIdx.x * 256 + threadIdx.x;
  float val = 0.f;
  if (tid < NP * NP) {
    int p = tid / NP, q = tid % NP;
    float s = 0.f;
    for (int t = 0; t < T_FRAMES; ++t)
      s += Dn[t * NP + p] * Dn[t * NP + q];
    DtD[tid] = s;                  // padded rows/cols are exact zeros
    val = s * s;
  }
  __shared__ float red[256];
  red[threadIdx.x] = val;
  __syncthreads();
  for (int s = 128; s > 0; s >>= 1) {
    if (threadIdx.x < s) red[threadIdx.x] += red[threadIdx.x + s];
    __syncthreads();
  }
  if (threadIdx.x == 0) atomicAdd(&scal[0], red[0]);
}

__global__ void k_finalize(float* __restrict__ scal) {
  float L = 1.f / sqrtf(scal[0]);  // L = 1/||DtD||_F
  scal[1] = L;
  scal[2] = L * LAMBDA_C;          // soft-threshold tau
}

// --------------------------------- DtY -------------------------------------
// DtY[b,p,j] = sum_t Dn[t,p] * y[b,t,j]   (K=64, cheap; coalesced over j)
__global__ __launch_bounds__(256)
void k_dty(const float* __restrict__ Dn, const float* __restrict__ y,
           float* __restrict__ DtY) {
  int j = blockIdx.x * 256 + threadIdx.x;
  int p = blockIdx.y, b = blockIdx.z;
  const float* yb = y + (long)b * T_FRAMES * NJ + j;
  float s = 0.f;
  for (int t = 0; t < T_FRAMES; ++t)
    s += Dn[t * NP + p] * yb[(long)t * NJ];
  DtY[((long)b * NP + p) * NJ + j] = s;
}

// ------------------- FISTA GEMM step (f32 WMMA) -----------------------------
// Cnext = soft_threshold(Cprev - L*(DtD @ Cprev - DtY), L*lambda)
// plus ||Cnext - Ccurr||^2 accumulation into norm2slot (for the tol gate).
// 6 waves as 3x2; each wave owns a 16x64 output strip = 4 WMMA accumulators.
__global__ __launch_bounds__(GEMM_THREADS)
void k_fista_gemm(const float* __restrict__ DtD, const float* __restrict__ DtY,
                  const float* __restrict__ Cprev, const float* __restrict__ Ccurr,
                  float* __restrict__ Cnext, const float* __restrict__ scal,
                  float* __restrict__ norm2slot) {
  __shared__ float As [BM][KP];    // A tile, row-major
  __shared__ float BsT[BN][KP];    // B tile TRANSPOSED: [n][k]; K-pair is 8B
  __shared__ float red[256];

  const int tid   = threadIdx.x;
  const int lane  = tid & 31;
  const int w     = tid >> 5;
  const int wr    = w >> 1;          // wave row 0..2  -> M offset 16*wr
  const int wc    = w & 1;           // wave col 0..1  -> N offset 64*wc
  const int khalf = lane >> 4;       // lane-half selects K pair / M half
  const int lrow  = lane & 15;

  const int  b  = blockIdx.z;
  const int  pm = blockIdx.y * BM;
  const int  jn = blockIdx.x * BN;
  const long cbase = (long)b * NP * NJ;

  const float Lv  = scal[1];
  const float thr = scal[2];

  v8f acc0 = {}, acc1 = {}, acc2 = {}, acc3 = {};

  const int arow = tid >> 2;         // 0..47 : A tile row
  const int akq  = (tid & 3) * 4;    // A tile 4-float chunk
  const int br0  = tid >> 5;         // B: row within K-tile (pass base)
  const int bc4  = (tid & 31) * 4;   // B: 4-float column chunk (coalesced)

  for (int kt = 0; kt < NP; kt += KT) {
    __syncthreads();
    // A tile: DtD[pm+row][kt..kt+15]  (48x16, one float4 per thread)
    {
      float4 va = *(const float4*)(DtD + (long)(pm + arow) * NP + kt + akq);
      As[arow][akq + 0] = va.x; As[arow][akq + 1] = va.y;
      As[arow][akq + 2] = va.z; As[arow][akq + 3] = va.w;
    }
    // B tile: Cprev[b][kt..kt+15][jn..jn+127] -> BsT[n][k]
    // 16 rows x 32 float4 = 512 chunks, 192 threads -> rows {br0, +6, +12}(<16)
#pragma unroll
    for (int pass = 0; pass < 3; ++pass) {
      int r0 = br0 + 6 * pass;
      if (r0 < KT) {
        float4 vb = *(const float4*)(Cprev + cbase + (long)(kt + r0) * NJ + jn + bc4);
        BsT[bc4 + 0][r0] = vb.x; BsT[bc4 + 1][r0] = vb.y;
        BsT[bc4 + 2][r0] = vb.z; BsT[bc4 + 3][r0] = vb.w;
      }
    }
    __syncthreads();

#pragma unroll
    for (int k0 = 0; k0 < KT; k0 += 4) {
      const int ka = k0 + 2 * khalf;   // lanes 0-15: K={k0,k0+1}; 16-31: {k0+2,k0+3}
      v2f a, b0, b1, b2, b3;
      a.x  = As[16 * wr + lrow][ka];     a.y  = As[16 * wr + lrow][ka + 1];
      const int nb = 64 * wc + lrow;
      b0.x = BsT[nb     ][ka];           b0.y = BsT[nb     ][ka + 1];
      b1.x = BsT[nb + 16][ka];           b1.y = BsT[nb + 16][ka + 1];
      b2.x = BsT[nb + 32][ka];           b2.y = BsT[nb + 32][ka + 1];
      b3.x = BsT[nb + 48][ka];           b3.y = BsT[nb + 48][ka + 1];
      acc0 = __builtin_amdgcn_wmma_f32_16x16x4_f32(false, a, false, b0,
                                                   (short)0, acc0, false, false);
      acc1 = __builtin_amdgcn_wmma_f32_16x16x4_f32(false, a, false, b1,
                                                   (short)0, acc1, false, false);
      acc2 = __builtin_amdgcn_wmma_f32_16x16x4_f32(false, a, false, b2,
                                                   (short)0, acc2, false, false);
      acc3 = __builtin_amdgcn_wmma_f32_16x16x4_f32(false, a, false, b3,
                                                   (short)0, acc3, false, false);
    }
  }

  // Fused FISTA epilogue. D layout: lane<16 -> M=v, lane>=16 -> M=8+v; N=lrow.
  float lsum = 0.f;
  const int prow = pm + 16 * wr + 8 * khalf;
#define EPILOGUE(ACC, NT)                                                     \
  {                                                                           \
    long idx = cbase + (long)p * NJ + jn + 64 * wc + 16 * (NT) + lrow;        \
    float g  = (ACC)[v] - DtY[idx];                                           \
    float cn = Cprev[idx] - Lv * g;                                           \
    float m  = fabsf(cn) - thr;                                               \
    cn = (m > 0.f) ? copysignf(m, cn) : 0.f;                                  \
    Cnext[idx] = cn;                                                          \
    float d = cn - Ccurr[idx];                                                \
    lsum += d * d;                                                            \
  }
#pragma unroll
  for (int v = 0; v < 8; ++v) {
    const int p = prow + v;
    EPILOGUE(acc0, 0)
    EPILOGUE(acc1, 1)
    EPILOGUE(acc2, 2)
    EPILOGUE(acc3, 3)
  }
#undef EPILOGUE
  red[tid] = lsum;
  if (tid < 64) red[192 + tid] = 0.f;   // pad 192 -> 256 for pow2 reduce
  __syncthreads();
  for (int s = 128; s > 0; s >>= 1) {
    if (tid < s) red[tid] += red[tid + s];
    __syncthreads();
  }
  if (tid == 0) atomicAdd(norm2slot, red[0]);
}

// -------------------------- momentum commit step ---------------------------
// frozen = sticky_flag | (||Cnext-Ccurr||^2 < tol^2); if frozen, state is
// kept exactly (matches reference "break before applying the update").
__global__ __launch_bounds__(256)
void k_commit(const float4* __restrict__ Cnext, float4* __restrict__ Cprev,
              float4* __restrict__ Ccurr, float* __restrict__ scal,
              const float* __restrict__ norm2slot, float tconst) {
  int frozen = (*(const int*)(scal + 3)) | (norm2slot[0] < TOL2 ? 1 : 0);
  if (frozen) {
    if (blockIdx.x == 0 && threadIdx.x == 0) *(int*)(scal + 3) = 1;
    return;
  }
  long i = (long)blockIdx.x * 256 + threadIdx.x;
  float4 cn = Cnext[i];
  float4 cc = Ccurr[i];
  float4 pn;
  pn.x = cn.x + tconst * (cn.x - cc.x);
  pn.y = cn.y + tconst * (cn.y - cc.y);
  pn.z = cn.z + tconst * (cn.z - cc.z);
  pn.w = cn.w + tconst * (cn.w - cc.w);
  Cprev[i] = pn;
  Ccurr[i] = cn;
}

// ------------------------------ reconstruction -----------------------------
// Y_pred[b,t,j] = sum_p Draw[t,p] * C[b,p,j]  (padded p terms are exact 0)
__global__ __launch_bounds__(256)
void k_recon(const float* __restrict__ Draw, const float* __restrict__ Ccurr,
             float* __restrict__ out) {
  int j = blockIdx.x * 256 + threadIdx.x;
  int t = blockIdx.y, b = blockIdx.z;
  const float* cb = Ccurr + (long)b * NP * NJ + j;
  float s = 0.f;
  for (int p = 0; p < NP; ++p)
    s += Draw[(long)t * NP + p] * cb[(long)p * NJ];
  out[((long)b * T_FRAMES + t) * NJ + j] = s;
}

// strip padding: C_pred output is [B, 513, 512]
__global__ __launch_bounds__(256)
void k_copy_c(const float* __restrict__ Ccurr, float* __restrict__ outC) {
  int j = blockIdx.x * 256 + threadIdx.x;
  int p = blockIdx.y, b = blockIdx.z;
  outC[((long)b * NP_TRUE + p) * NJ + j] = Ccurr[((long)b * NP + p) * NJ + j];
}

// ------------------------------- launcher ----------------------------------
extern "C" void kernel_launch(void* const* d_in, const int* in_sizes, int n_in,
                              void* d_out, int out_size, void* d_ws, size_t ws_size,
                              hipStream_t stream) {
  (void)in_sizes; (void)n_in; (void)out_size; (void)ws_size;
  const float* y  = (const float*)d_in[0];   // [32,64,512]
  const float* r  = (const float*)d_in[1];   // [256]
  const float* th = (const float*)d_in[2];   // [256]
  float* out = (float*)d_out;                // Y_pred (1,048,576) ++ C_pred (8,404,992)

  // workspace layout (floats, all 16B aligned); ~140 MB total, L2-resident
  float* ws = (float*)d_ws;
  size_t off = 0;
  float* Draw = ws + off; off += (size_t)T_FRAMES * NP;   // un-normalized dict
  float* Dn   = ws + off; off += (size_t)T_FRAMES * NP;   // normalized dict
  float* DtD  = ws + off; off += (size_t)NP * NP;
  float* scal = ws + off; off += 64;   // [0]=fro2 [1]=L [2]=tau [3]=flag [4..33]=norm2
  const size_t CSZ = (size_t)NBATCH * NP * NJ;            // 8,650,752
  float* DtY   = ws + off; off += CSZ;
  float* Cprev = ws + off; off += CSZ;                    // contiguous with Ccurr
  float* Ccurr = ws + off; off += CSZ;
  float* Cnext = ws + off; off += CSZ;

  k_build_dict<<<1, 544, 0, stream>>>(r, th, Draw, Dn, scal);
  {
    long n4 = (long)(2 * CSZ) / 4;                        // zero Cprev + Ccurr
    k_zero4<<<(int)((n4 + 255) / 256), 256, 0, stream>>>((float4*)Cprev, n4);
  }
  k_dtd<<<(NP * NP + 255) / 256, 256, 0, stream>>>(Dn, DtD, scal);
  k_finalize<<<1, 1, 0, stream>>>(scal);
  k_dty<<<dim3(NJ / 256, NP, NBATCH), 256, 0, stream>>>(Dn, y, DtY);

  double tp = 1.0;  // FISTA t-sequence is data-independent: precompute t_const
  for (int it = 0; it < NITER; ++it) {
    double tn = (1.0 + sqrt(1.0 + 4.0 * tp * tp)) * 0.5;
    float tconst = (float)((tp - 1.0) / tn);
    k_fista_gemm<<<dim3(NJ / BN, NP / BM, NBATCH), GEMM_THREADS, 0, stream>>>(
        DtD, DtY, Cprev, Ccurr, Cnext, scal, scal + 4 + it);
    k_commit<<<(int)(CSZ / 4 / 256), 256, 0, stream>>>(
        (const float4*)Cnext, (float4*)Cprev, (float4*)Ccurr, scal,
        scal + 4 + it, tconst);
    tp = tn;
  }

  k_recon<<<dim3(NJ / 256, T_FRAMES, NBATCH), 256, 0, stream>>>(Draw, Ccurr, out);
  k_copy_c<<<dim3(NJ / 256, NP_TRUE, NBATCH), 256, 0, stream>>>(
      Ccurr, out + (size_t)NBATCH * T_FRAMES * NJ);
}